// SageLayer_73409581023297
// MI455X (gfx1250) — compile-verified
//
#include <hip/hip_runtime.h>

#define NODE_DIM 64

typedef float v2f __attribute__((ext_vector_type(2)));
typedef float v8f __attribute__((ext_vector_type(8)));

// ---------------- zero workspace (neigh accum + deg, contiguous) ----------------
__global__ void sage_zero_kernel(float* __restrict__ ws, long long n) {
  long long i = (long long)blockIdx.x * blockDim.x + threadIdx.x;
  if (i < n) ws[i] = 0.0f;
}

// ---------------- edge scatter: neigh[row] += features[col]; deg[row] += 1 -----
// 16 threads per edge, each handles 4 contiguous floats (float4 gather,
// 4x global_atomic_add_f32 scatter). Accumulator + features are L2-resident.
__global__ void sage_scatter_kernel(const float* __restrict__ feat,
                                    const int* __restrict__ row,
                                    const int* __restrict__ col,
                                    float* __restrict__ neigh,
                                    float* __restrict__ deg,
                                    int n_edges) {
  long long tid = (long long)blockIdx.x * blockDim.x + threadIdx.x;
  long long e   = tid >> 4;
  int c4        = ((int)tid & 15) * 4;
  if (e >= n_edges) return;
  int r = row[e];
  int c = col[e];
  const float4 v = *(const float4*)(feat + (long long)c * NODE_DIM + c4);
  float* dst = neigh + (long long)r * NODE_DIM + c4;
  atomicAdd(dst + 0, v.x);
  atomicAdd(dst + 1, v.y);
  atomicAdd(dst + 2, v.z);
  atomicAdd(dst + 3, v.w);
  if (c4 == 0) atomicAdd(deg + r, 1.0f);
}

// ---------------- GEMM + ReLU via V_WMMA_F32_16X16X4_F32 -----------------------
// out[m][n] = relu( sum_k X[m][k] * W[k][n] ), X = concat(feat, neigh/(deg+1)).
// 256-thread block = 8 waves; W (64 KB) staged once into LDS, k-pair interleaved:
//   sW2[(k/2)*256 + n*2 + (k&1)] = W[k][n]
// so each B operand {W[k][n], W[k+1][n]} is one aligned ds_load_b64 (no repack).
// Each wave owns one 16-row slab and computes all 8 column tiles (16x128)
// with 8 independent f32-WMMA accumulator chains; deg-normalization fused
// into the A load for K >= 64.
// A layout (16x4 f32): lane = M (mod 16), vgpr pair = {K,K+1}; lanes 16-31 = K+2/K+3.
// B layout (4x16 f32): lane = N (mod 16), same K split.
// C/D layout: lane = N, element v -> row M = v + 8*(lane>=16).
__global__ __launch_bounds__(256) void sage_gemm_relu_kernel(
    const float* __restrict__ feat, const float* __restrict__ neigh,
    const float* __restrict__ deg, const float* __restrict__ W,
    float* __restrict__ out, int n_nodes) {
  __shared__ float sW2[128 * 128];  // 64 KB, k-pair interleaved

  // Cooperative swizzled stage of W: per chunk, read 4 cols of rows {2kp, 2kp+1}
  // (two b128 global loads) and write interleaved pairs (two b128 LDS stores).
  {
    const int t = threadIdx.x;
#pragma unroll
    for (int i = 0; i < 8; ++i) {
      const int j  = i * 256 + t;     // 2048 chunks: kp = j>>5, n4 = (j&31)*4
      const int kp = j >> 5;
      const int n4 = (j & 31) * 4;
      const float4 r0 = *(const float4*)(W + (2 * kp)     * 128 + n4);
      const float4 r1 = *(const float4*)(W + (2 * kp + 1) * 128 + n4);
      float* dst = sW2 + kp * 256 + n4 * 2;
      const float4 s0 = {r0.x, r1.x, r0.y, r1.y};
      const float4 s1 = {r0.z, r1.z, r0.w, r1.w};
      *(float4*)(dst + 0) = s0;
      *(float4*)(dst + 4) = s1;
    }
  }
  __syncthreads();

  const int lane = threadIdx.x & 31;
  const int wave = threadIdx.x >> 5;
  const int half = lane >> 4;      // 0: K+{0,1}, 1: K+{2,3}
  const int lm   = lane & 15;
  const int tile_m = blockIdx.x * 8 + wave;

  int m = tile_m * 16 + lm;
  if (m >= n_nodes) m = n_nodes - 1;  // clamp loads: WMMA requires EXEC all-ones
  const int koff = half * 2;

  const float* __restrict__ arow  = feat  + (long long)m * NODE_DIM;
  const float* __restrict__ arow2 = neigh + (long long)m * NODE_DIM;
  const float inv = 1.0f / (deg[m] + 1.0f);   // fused normalization factor

  v8f acc[8] = {};
#pragma unroll
  for (int k0 = 0; k0 < 128; k0 += 4) {
    const int ka = k0 + koff;
    v2f a;
    if (k0 < 64) {
      a.x = arow[ka];
      a.y = arow[ka + 1];
    } else {
      a.x = arow2[ka - 64] * inv;
      a.y = arow2[ka - 63] * inv;
    }
    // B pairs {W[ka][n], W[ka+1][n]} are contiguous b64s in LDS.
    const v2f* wrow = (const v2f*)sW2 + (k0 / 2 + half) * 128 + lm;
#pragma unroll
    for (int t = 0; t < 8; ++t) {
      const v2f b = wrow[t * 16];
      acc[t] = __builtin_amdgcn_wmma_f32_16x16x4_f32(false, a, false, b,
                                                     (short)0, acc[t],
                                                     false, false);
    }
  }

  const int mbase = tile_m * 16 + half * 8;
#pragma unroll
  for (int v = 0; v < 8; ++v) {
    const int mo = mbase + v;
    if (mo < n_nodes) {
      float* __restrict__ orow = out + (long long)mo * 128 + lm;
#pragma unroll
      for (int t = 0; t < 8; ++t) {
        const float val = acc[t][v];
        orow[t * 16] = val > 0.0f ? val : 0.0f;
      }
    }
  }
}

extern "C" void kernel_launch(void* const* d_in, const int* in_sizes, int n_in,
                              void* d_out, int out_size, void* d_ws, size_t ws_size,
                              hipStream_t stream) {
  const float* feat = (const float*)d_in[0];   // [N, 64] f32
  const int*   row  = (const int*)d_in[1];     // [E]
  const int*   col  = (const int*)d_in[2];     // [E]
  const float* W    = (const float*)d_in[3];   // [128, 128] f32
  float*       out  = (float*)d_out;           // [N, 128] f32

  const int n_nodes = in_sizes[0] / NODE_DIM;
  const int n_edges = in_sizes[1];

  // workspace partition: neigh accumulator then degree (contiguous for zeroing)
  float* neigh = (float*)d_ws;                              // N*64 floats
  float* deg   = neigh + (long long)n_nodes * NODE_DIM;     // N floats

  const long long zn = (long long)n_nodes * (NODE_DIM + 1);
  sage_zero_kernel<<<(int)((zn + 255) / 256), 256, 0, stream>>>(neigh, zn);

  const long long st = (long long)n_edges * 16;
  sage_scatter_kernel<<<(int)((st + 255) / 256), 256, 0, stream>>>(
      feat, row, col, neigh, deg, n_edges);

  const int ntiles = (n_nodes + 15) / 16;
  sage_gemm_relu_kernel<<<(ntiles + 7) / 8, 256, 0, stream>>>(
      feat, neigh, deg, W, out, n_nodes);
}